// _FallbackSSM_68874095559514
// MI455X (gfx1250) — compile-verified
//
#include <hip/hip_runtime.h>
#include <hip/hip_bf16.h>
#include <math.h>

// ---------------------------------------------------------------------------
// Mamba SSM block for MI455X (gfx1250, wave32).
// GEMMs on V_WMMA_F32_16X16X4_F32 (native fp32 matrix pipe): matches the fp32
// reference numerics. Exact tiling (templated) -> zero divergence around WMMA
// (EXEC all-ones), straight-line inner loop. Wave tile 32x64: B-frags reused
// across two M-tiles in registers -> 8 WMMAs per 10 vmem issues.
// Working set ~150MB -> L2-resident (192MB global L2).
// ---------------------------------------------------------------------------

typedef float v2f __attribute__((ext_vector_type(2)));
typedef float v8f __attribute__((ext_vector_type(8)));

#define D_MODEL 768
#define D_STATE 16
#define D_CONV  4
#define D_INNER 1536
#define BATCH   2
#define SEQ     2048
#define NTOK    (BATCH * SEQ)   // 4096 tokens

__device__ __forceinline__ float silu_f(float x) {
    return x / (1.0f + __expf(-x));
}
__device__ __forceinline__ float softplus_f(float x) {
    return (x > 20.0f) ? x : __logf(1.0f + __expf(x));
}

// ---------------------------------------------------------------------------
// C[M,N] = A[M,K] * B[K,N]   (row-major, fp32, WMMA 16x16x4)
// Wave tile: 32(M) x 16*JT(N)  (2 M-tiles share the JT B-frags).
// Block: 128 threads = 4 waves stacked along M -> block tile 128 x 16*JT.
// Requires: M % 128 == 0, N % (16*JT) == 0, K % 8 == 0. No predicates.
//
// A-frag (16x4 MxK, ISA 7.12.2): m = lane&15; one float2 = K {kk, kk+1},
//   kk = k + 2*(lane>>4).
// B-frag (4x16 KxN): n = n0 + 16j + (lane&15), rows K mirrored like A.
// C/D: elem i of v8f at lane l -> M = i + 8*(l>>4), N = l&15.
// ---------------------------------------------------------------------------
template <int JT>
__global__ void __launch_bounds__(128) gemm_f32_wmma(
    const float* __restrict__ A, const float* __restrict__ Bm,
    float* __restrict__ C,
    int K, int lda, int ldb, int ldc)
{
    const int lane = threadIdx.x & 31;
    const int wave = threadIdx.x >> 5;
    const int m0   = blockIdx.y * 128 + wave * 32;   // two 16-row tiles
    const int n0   = blockIdx.x * (16 * JT);
    const int l16  = lane & 15;
    const int hi   = lane >> 4;       // 0 or 1

    v8f acc0[JT] = {};
    v8f acc1[JT] = {};

    const float* Arow0 = A + (size_t)(m0 + l16) * lda;
    const float* Arow1 = Arow0 + (size_t)16 * lda;
    const float* Bcol  = Bm + n0 + l16;

#pragma unroll 2
    for (int k = 0; k < K; k += 4) {
        const int kk = k + 2 * hi;
        const v2f a0 = *reinterpret_cast<const v2f*>(Arow0 + kk);
        const v2f a1 = *reinterpret_cast<const v2f*>(Arow1 + kk);
        v2f b[JT];
#pragma unroll
        for (int j = 0; j < JT; ++j) {
            b[j].x = Bcol[(size_t)kk * ldb + j * 16];
            b[j].y = Bcol[(size_t)(kk + 1) * ldb + j * 16];
        }
#pragma unroll
        for (int j = 0; j < JT; ++j) {
            acc0[j] = __builtin_amdgcn_wmma_f32_16x16x4_f32(
                false, a0, false, b[j], (short)0, acc0[j], false, false);
        }
#pragma unroll
        for (int j = 0; j < JT; ++j) {
            acc1[j] = __builtin_amdgcn_wmma_f32_16x16x4_f32(
                false, a1, false, b[j], (short)0, acc1[j], false, false);
        }
    }

#pragma unroll
    for (int j = 0; j < JT; ++j) {
        const int ncol = n0 + j * 16 + l16;
#pragma unroll
        for (int i = 0; i < 8; ++i) {
            const int mr = m0 + i + 8 * hi;
            C[(size_t)mr * ldc + ncol]        = acc0[j][i];
            C[(size_t)(mr + 16) * ldc + ncol] = acc1[j][i];
        }
    }
}

// ---------------------------------------------------------------------------
// Depthwise causal conv (kernel 4) over time + bias + SiLU.
// x_inner lives in xz[..., :D_INNER] (row stride 2*D_INNER).
// ---------------------------------------------------------------------------
__global__ void __launch_bounds__(256) conv_silu_kernel(
    const float* __restrict__ xz, const float* __restrict__ Wc,
    const float* __restrict__ bc, float* __restrict__ xc)
{
    const int idx = blockIdx.x * blockDim.x + threadIdx.x;
    if (idx >= NTOK * D_INNER) return;
    const int d = idx % D_INNER;
    const int t = idx / D_INNER;       // flattened b*L + l
    const int l = t % SEQ;

    float s = bc[d];
#pragma unroll
    for (int j = 0; j < D_CONV; ++j) {
        const int ls = l - (D_CONV - 1) + j;
        if (ls >= 0)
            s += xz[(size_t)(t - (D_CONV - 1) + j) * (2 * D_INNER) + d] *
                 Wc[d * D_CONV + j];
    }
    xc[idx] = silu_f(s);
}

// ---------------------------------------------------------------------------
// Selective scan. One thread per (b, d, n) state -> 49152 threads (1536 waves)
// hide the 2048-step sequential latency. 16-lane shuffle tree reduces over n.
// ---------------------------------------------------------------------------
__global__ void __launch_bounds__(256) ssm_scan_kernel(
    const float* __restrict__ xz,    // z slice: xz[..., D_INNER:]
    const float* __restrict__ xc,
    const float* __restrict__ dtp,   // xc @ W_dt (pre bias/softplus)
    const float* __restrict__ bdt,
    const float* __restrict__ BCb,   // [NTOK, 2*D_STATE]
    const float* __restrict__ A_log,
    const float* __restrict__ Dp,
    float* __restrict__ yf)
{
    const int n    = threadIdx.x & 15;
    const int dl   = threadIdx.x >> 4;
    const int pair = blockIdx.x * (256 / 16) + dl;   // b*D_INNER + d
    const int b    = pair / D_INNER;
    const int d    = pair % D_INNER;

    const float Aval = -__expf(A_log[d * D_STATE + n]);
    const float bdtv = bdt[d];
    const float Dv   = Dp[d];

    float h = 0.0f;
    for (int l = 0; l < SEQ; ++l) {
        const size_t t = (size_t)b * SEQ + l;
        const float dtv = softplus_f(dtp[t * D_INNER + d] + bdtv);
        const float xcv = xc[t * D_INNER + d];
        const float Bv  = BCb[t * (2 * D_STATE) + n];
        const float Cv  = BCb[t * (2 * D_STATE) + D_STATE + n];

        h = __expf(dtv * Aval) * h + dtv * Bv * xcv;

        float yp = h * Cv;                 // reduce over n (16 lanes)
        yp += __shfl_xor(yp, 8, 16);
        yp += __shfl_xor(yp, 4, 16);
        yp += __shfl_xor(yp, 2, 16);
        yp += __shfl_xor(yp, 1, 16);

        if (n == 0) {
            const float zv = xz[t * (2 * D_INNER) + D_INNER + d];
            yf[t * D_INNER + d] = (yp + xcv * Dv) * silu_f(zv);
        }
    }
}

// ---------------------------------------------------------------------------
extern "C" void kernel_launch(void* const* d_in, const int* in_sizes, int n_in,
                              void* d_out, int out_size, void* d_ws, size_t ws_size,
                              hipStream_t stream)
{
    const float* x      = (const float*)d_in[0];
    const float* W_in   = (const float*)d_in[1];
    const float* W_conv = (const float*)d_in[2];
    const float* b_conv = (const float*)d_in[3];
    const float* W_x    = (const float*)d_in[4];
    const float* W_dt   = (const float*)d_in[5];
    const float* b_dt   = (const float*)d_in[6];
    const float* A_log  = (const float*)d_in[7];
    const float* Dp     = (const float*)d_in[8];
    const float* W_out  = (const float*)d_in[9];
    float* out = (float*)d_out;

    float* ws  = (float*)d_ws;
    float* xz  = ws;                                  // [NTOK, 3072]
    float* xc  = xz  + (size_t)NTOK * 2 * D_INNER;    // [NTOK, 1536]
    float* dtp = xc  + (size_t)NTOK * D_INNER;        // [NTOK, 1536]
    float* BCb = dtp + (size_t)NTOK * D_INNER;        // [NTOK, 32]
    float* yf  = BCb + (size_t)NTOK * 2 * D_STATE;    // [NTOK, 1536]

    const dim3 blk(128);
    const int mblocks = NTOK / 128;   // 32

    // 1) xz = x @ W_in            (4096 x 3072 x 768), JT=4: grid 48 x 32
    gemm_f32_wmma<4><<<dim3((2 * D_INNER) / 64, mblocks), blk, 0, stream>>>(
        x, W_in, xz, D_MODEL, D_MODEL, 2 * D_INNER, 2 * D_INNER);

    // 2) xc = silu(depthwise_conv(x_inner) + b_conv)
    const int total = NTOK * D_INNER;
    conv_silu_kernel<<<(total + 255) / 256, 256, 0, stream>>>(xz, W_conv, b_conv, xc);

    // 3) BC = xc @ W_x            (4096 x 32 x 1536), JT=2: grid 1 x 32
    gemm_f32_wmma<2><<<dim3((2 * D_STATE) / 32, mblocks), blk, 0, stream>>>(
        xc, W_x, BCb, D_INNER, D_INNER, 2 * D_STATE, 2 * D_STATE);

    // 4) dtp = xc @ W_dt          (4096 x 1536 x 1536), JT=4: grid 24 x 32
    gemm_f32_wmma<4><<<dim3(D_INNER / 64, mblocks), blk, 0, stream>>>(
        xc, W_dt, dtp, D_INNER, D_INNER, D_INNER, D_INNER);

    // 5) selective scan -> yf
    ssm_scan_kernel<<<BATCH * D_INNER / 16, 256, 0, stream>>>(
        xz, xc, dtp, b_dt, BCb, A_log, Dp, yf);

    // 6) out = yf @ W_out         (4096 x 768 x 1536), JT=4: grid 12 x 32
    gemm_f32_wmma<4><<<dim3(D_MODEL / 64, mblocks), blk, 0, stream>>>(
        yf, W_out, out, D_INNER, D_INNER, D_MODEL, D_MODEL);
}